// Spalize_78426102825319
// MI455X (gfx1250) — compile-verified
//
#include <hip/hip_runtime.h>
#include <hip/hip_bf16.h>

// spalize: out[k,c,h,w] = img[c,h,w] * (mask[h,w]==k), plus mask passthrough.
// Pure store-bandwidth kernel: 157MB of b128 NT stores; inputs staged to LDS
// via the gfx1250 async-to-LDS DMA path (ASYNCcnt), issued with inline asm.

typedef __attribute__((ext_vector_type(4))) float v4f;
typedef __attribute__((ext_vector_type(4))) int   v4i;

#define TILE     1024   // pixels per block
#define KPB      5      // k-slices per block
#define NTHREADS 256    // 8 wave32s

#if defined(__AMDGCN__)
#define HAVE_ASYNC_LDS 1
#else
#define HAVE_ASYNC_LDS 0
#endif

#if HAVE_ASYNC_LDS
// Copy 16 bytes global->LDS asynchronously (tracked by ASYNCcnt).
// VGLOBAL async encoding: VDST = LDS byte address VGPR, VADDR = 64-bit global
// address VGPR pair, SADDR = off.
__device__ __forceinline__ void async_copy_b128(const void* gsrc, void* ldst) {
    __attribute__((address_space(3))) void* l =
        (__attribute__((address_space(3))) void*)ldst;          // 32-bit LDS addr
    const __attribute__((address_space(1))) void* g =
        (const __attribute__((address_space(1))) void*)gsrc;    // 64-bit global addr
    asm volatile("global_load_async_to_lds_b128 %0, %1, off"
                 :
                 : "v"(l), "v"(g)
                 : "memory");
}
#endif

__global__ __launch_bounds__(NTHREADS)
void spalize_scatter_kernel(const float* __restrict__ img,
                            const int*   __restrict__ mask,
                            float* __restrict__ out,
                            int HW, int K)
{
    __shared__ float s_img[3 * TILE];
    __shared__ int   s_mask[TILE];

    const int tid      = threadIdx.x;
    const int tileBase = blockIdx.x * TILE;
    const int kBase    = blockIdx.y * KPB;
    const int p4       = tid * 4;        // 4 pixels / thread, 16B aligned

#if HAVE_ASYNC_LDS
    async_copy_b128(mask + tileBase + p4, &s_mask[p4]);
    #pragma unroll
    for (int c = 0; c < 3; ++c) {
        async_copy_b128(img + c * HW + tileBase + p4, &s_img[c * TILE + p4]);
    }
#if __has_builtin(__builtin_amdgcn_s_wait_asynccnt)
    __builtin_amdgcn_s_wait_asynccnt(0);
#else
    asm volatile("s_wait_asynccnt 0" ::: "memory");
#endif
#else
    // Host-pass / fallback staging (b128 load + ds_store_b128).
    {
        v4i mv = *(const v4i*)(mask + tileBase + p4);
        *(v4i*)&s_mask[p4] = mv;
        #pragma unroll
        for (int c = 0; c < 3; ++c) {
            v4f iv = *(const v4f*)(img + c * HW + tileBase + p4);
            *(v4f*)&s_img[c * TILE + p4] = iv;
        }
    }
#endif
    __syncthreads();

    const v4i m = *(const v4i*)&s_mask[p4];   // ds_load_b128

    #pragma unroll
    for (int kk = 0; kk < KPB; ++kk) {
        const int k = kBase + kk;
        if (k >= K) break;
        #pragma unroll
        for (int c = 0; c < 3; ++c) {
            const v4f iv = *(const v4f*)&s_img[c * TILE + p4];
            v4f r;
            r.x = (m.x == k) ? iv.x : 0.0f;
            r.y = (m.y == k) ? iv.y : 0.0f;
            r.z = (m.z == k) ? iv.z : 0.0f;
            r.w = (m.w == k) ? iv.w : 0.0f;
            float* dst = out + (size_t)(k * 3 + c) * (size_t)HW + tileBase + p4;
            __builtin_nontemporal_store(r, (v4f*)dst);  // global_store_b128 th:NT
        }
    }

    // Second tuple element: mask passthrough (as the output's float dtype).
    if (blockIdx.y == 0) {
        v4f fm;
        fm.x = (float)m.x; fm.y = (float)m.y;
        fm.z = (float)m.z; fm.w = (float)m.w;
        float* dst = out + (size_t)K * 3 * (size_t)HW + tileBase + p4;
        __builtin_nontemporal_store(fm, (v4f*)dst);
    }
}

extern "C" void kernel_launch(void* const* d_in, const int* in_sizes, int n_in,
                              void* d_out, int out_size, void* d_ws, size_t ws_size,
                              hipStream_t stream) {
    const float* img  = (const float*)d_in[0];   // [3, 512, 512] f32
    const int*   mask = (const int*)d_in[1];     // [512, 512] i32 (jax default x64-off)
    float*       out  = (float*)d_out;           // [K,3,H,W] f32 ++ [H,W] mask-as-float

    const int HW = in_sizes[1];                        // 262144
    const int C  = in_sizes[0] / HW;                   // 3 (kernel assumes 3)
    const int K  = (out_size - HW) / in_sizes[0];      // 50
    (void)C; (void)n_in; (void)d_ws; (void)ws_size;

    const int tiles   = HW / TILE;                     // 256
    const int kgroups = (K + KPB - 1) / KPB;           // 10

    dim3 grid(tiles, kgroups, 1);
    spalize_scatter_kernel<<<grid, NTHREADS, 0, stream>>>(img, mask, out, HW, K);
}